// TokenMixer_64836826300527
// MI455X (gfx1250) — compile-verified
//
#include <hip/hip_runtime.h>
#include <stdint.h>

// ---------------------------------------------------------------------------
// TokenMixer: out = x_pos * (1 - sigmoid(cos_sim(x_pre, x_pos, axis=C)))
// Shapes: [N=4096, B=16, C=512] fp32 -> 65536 independent rows of 512 floats.
// Memory-bound (403 MB traffic, ~0.27 GFLOP): stream via CDNA5 Tensor Data
// Mover (async DMA tiles -> LDS, TENSORcnt) + non-temporal b128 stores.
// ---------------------------------------------------------------------------

typedef __attribute__((ext_vector_type(4))) unsigned int v4u;
typedef __attribute__((ext_vector_type(8))) int          v8i;
typedef __attribute__((ext_vector_type(4))) int          v4i;
typedef __attribute__((ext_vector_type(4))) float        f4;

#define ROW_C          512
#define ROWS_PER_TILE  8
#define TILE_ELEMS     (ROW_C * ROWS_PER_TILE)   // 4096 floats
#define TILE_BYTES     (TILE_ELEMS * 4)          // 16 KiB

// Issue one TDM DMA: copy a contiguous 16 KiB tile (4096 fp32) from global
// memory to LDS. Builds the Tensor DMA Descriptor (D#) per CDNA5 ISA ch.8.
__device__ __forceinline__ void tdm_load_tile(uint32_t lds_byte_addr,
                                              const float* gptr) {
  const uint64_t ga = (uint64_t)(uintptr_t)gptr;

  // D# group 0 (128b): [1:0]=count=1, [63:32]=lds_addr, [120:64]=global_addr,
  // [127:126]=type=2 ("image").
  v4u g0;
  g0.x = 1u;
  g0.y = lds_byte_addr;
  g0.z = (uint32_t)(ga & 0xFFFFFFFFu);
  g0.w = (uint32_t)((ga >> 32) & 0x01FFFFFFu) | (2u << 30);

  // D# group 1 (256b): workgroup_mask=0 (not in cluster), data_size=2 (4B),
  // no atomic-barrier / iterate / pad. 1-D tile: tile_dim0 = 4096 elements,
  // tensor_dim0 = 4096 (tile fully in bounds), higher dims unused.
  const uint32_t d0 = TILE_ELEMS;     // tensor_dim0 (bits 79:48)
  const uint32_t d1 = 1u;             // tensor_dim1 (bits 111:80)
  const uint32_t t0 = TILE_ELEMS;     // tile_dim0   (bits 127:112)
  const uint64_t s0 = TILE_ELEMS;     // tensor_dim0_stride (bits 207:160)
  v8i g1;
  g1[0] = (int)(2u << 16);                                        // data_size=4B
  g1[1] = (int)((d0 & 0xFFFFu) << 16);                            // dim0 lo
  g1[2] = (int)(((d0 >> 16) & 0xFFFFu) | ((d1 & 0xFFFFu) << 16)); // dim0 hi|dim1 lo
  g1[3] = (int)(((d1 >> 16) & 0xFFFFu) | ((t0 & 0xFFFFu) << 16)); // dim1 hi|tile0
  g1[4] = 0;                                                      // tile1=0, tile2=0
  g1[5] = (int)(uint32_t)(s0 & 0xFFFFFFFFu);                      // stride0 lo
  g1[6] = (int)(uint32_t)((s0 >> 32) & 0xFFFFu);                  // stride0 hi
  g1[7] = 0;

  const v4i gz4 = {0, 0, 0, 0};              // groups 2/3 unused (<=2D tensor)
  const v8i gz8 = {0, 0, 0, 0, 0, 0, 0, 0};  // trailing group (clang-23 form)
  // amdgpu-toolchain (clang-23) 6-arg form: (g0, g1, g2, g3, gX, cpol)
  __builtin_amdgcn_tensor_load_to_lds(g0, g1, gz4, gz4, gz8, 0);
}

__global__ void __launch_bounds__(256)
tokenmixer_kernel(const float* __restrict__ x_pre,
                  const float* __restrict__ x_pos,
                  float* __restrict__ out,
                  int totalTiles, int tilesPerBlock) {
  // [buf(2)][mat(2)][4096 floats] = 64 KiB, double-buffered TDM staging.
  __shared__ __align__(16) float lds[2 * 2 * TILE_ELEMS];
  const uint32_t ldsBase = (uint32_t)(uintptr_t)&lds[0];  // low 32b = LDS offset

  const int tid      = threadIdx.x;
  const int lane     = tid & 31;
  const int wave     = tid >> 5;                 // 0..7, one row per wave
  const int tileBase = blockIdx.x * tilesPerBlock;

  // ---- Pipeline prologue: wave 0 issues DMA for the first two tiles. ----
  if (tid < 32) {
    if (tileBase < totalTiles) {
      tdm_load_tile(ldsBase + 0 * TILE_BYTES, x_pre + (size_t)tileBase * TILE_ELEMS);
      tdm_load_tile(ldsBase + 1 * TILE_BYTES, x_pos + (size_t)tileBase * TILE_ELEMS);
    }
    if (tilesPerBlock > 1 && tileBase + 1 < totalTiles) {
      tdm_load_tile(ldsBase + 2 * TILE_BYTES, x_pre + (size_t)(tileBase + 1) * TILE_ELEMS);
      tdm_load_tile(ldsBase + 3 * TILE_BYTES, x_pos + (size_t)(tileBase + 1) * TILE_ELEMS);
    }
  }

  for (int t = 0; t < tilesPerBlock; ++t) {
    const int tile = tileBase + t;
    const int buf  = t & 1;

    // TDM ops from one wave complete in order: waiting to <=2 outstanding
    // guarantees this tile's two loads have landed (only tile t+1's remain).
    if (tid < 32) {
      if (t + 1 < tilesPerBlock && tile + 1 < totalTiles)
        __builtin_amdgcn_s_wait_tensorcnt(2);
      else
        __builtin_amdgcn_s_wait_tensorcnt(0);
    }
    __syncthreads();   // tile data in lds[buf] now visible to all 8 waves

    if (tile < totalTiles) {
      const float* pre = &lds[(buf * 2 + 0) * TILE_ELEMS + wave * ROW_C];
      const float* pos = &lds[(buf * 2 + 1) * TILE_ELEMS + wave * ROW_C];

      f4 b[4];
      float saa = 0.f, sab = 0.f, sbb = 0.f;
#pragma unroll
      for (int j = 0; j < 4; ++j) {
        const int idx = (j * 32 + lane) * 4;        // strided: conflict-free b128
        const f4 a = *(const f4*)(pre + idx);
        b[j]       = *(const f4*)(pos + idx);
        saa += a.x * a.x + a.y * a.y + a.z * a.z + a.w * a.w;
        sbb += b[j].x * b[j].x + b[j].y * b[j].y + b[j].z * b[j].z + b[j].w * b[j].w;
        sab += a.x * b[j].x + a.y * b[j].y + a.z * b[j].z + a.w * b[j].w;
      }

      // wave32 tree reduction of the three partials
#pragma unroll
      for (int s = 16; s > 0; s >>= 1) {
        saa += __shfl_xor(saa, s, 32);
        sab += __shfl_xor(sab, s, 32);
        sbb += __shfl_xor(sbb, s, 32);
      }

      const float eps = 1e-12f;                     // F.normalize eps
      const float na  = fmaxf(sqrtf(saa), eps);
      const float nb  = fmaxf(sqrtf(sbb), eps);
      const float dis = sab / (na * nb);
      const float w   = 1.0f / (1.0f + expf(dis));  // 1 - sigmoid(dis)

      float* op = out + (size_t)tile * TILE_ELEMS + wave * ROW_C;
#pragma unroll
      for (int j = 0; j < 4; ++j) {
        const int idx = (j * 32 + lane) * 4;
        f4 o;
        o.x = b[j].x * w; o.y = b[j].y * w; o.z = b[j].z * w; o.w = b[j].w * w;
        // Streaming store: 403 MB > 192 MB L2, touched once -> non-temporal.
        __builtin_nontemporal_store(o, (f4*)(op + idx));
      }
    }
    __syncthreads();   // all waves done reading lds[buf] before DMA reuses it

    // ---- Pipeline steady state: refill this buffer with tile t+2. ----
    if (tid < 32 && t + 2 < tilesPerBlock && tile + 2 < totalTiles) {
      tdm_load_tile(ldsBase + (buf * 2 + 0) * TILE_BYTES,
                    x_pre + (size_t)(tile + 2) * TILE_ELEMS);
      tdm_load_tile(ldsBase + (buf * 2 + 1) * TILE_BYTES,
                    x_pos + (size_t)(tile + 2) * TILE_ELEMS);
    }
  }
}

extern "C" void kernel_launch(void* const* d_in, const int* in_sizes, int n_in,
                              void* d_out, int out_size, void* d_ws, size_t ws_size,
                              hipStream_t stream) {
  const float* x_pre = (const float*)d_in[0];
  const float* x_pos = (const float*)d_in[1];
  float* out = (float*)d_out;

  const int totalElems = in_sizes[0];                        // 4096*16*512
  const int rows       = totalElems / ROW_C;                 // 65536
  const int tiles      = (rows + ROWS_PER_TILE - 1) / ROWS_PER_TILE;  // 8192
  int blocks = tiles < 1024 ? tiles : 1024;
  int tpb    = (tiles + blocks - 1) / blocks;                // 8

  tokenmixer_kernel<<<blocks, 256, 0, stream>>>(x_pre, x_pos, out, tiles, tpb);
}